// GCNLayer_49916109914170
// MI455X (gfx1250) — compile-verified
//
#include <hip/hip_runtime.h>
#include <hip/hip_bf16.h>

typedef float v2f __attribute__((ext_vector_type(2)));
typedef float v8f __attribute__((ext_vector_type(8)));

#define D_FEAT 128

__device__ __forceinline__ void atomAddF(float* p, float v) {
#if defined(__HIP_DEVICE_COMPILE__)
    unsafeAtomicAdd(p, v);   // lowers to global_atomic_add_f32 on gfx1250
#else
    atomicAdd(p, v);
#endif
}

// ---------------------------------------------------------------------------
// 1) init: agg = 0, deg = 1.0f (self loop included in degree)
// ---------------------------------------------------------------------------
__global__ __launch_bounds__(256) void gcn_init(float4* __restrict__ agg4,
                                                float* __restrict__ deg, int N) {
    int i = blockIdx.x * 256 + threadIdx.x;
    int n4 = N * (D_FEAT / 4);
    if (i < n4) agg4[i] = make_float4(0.f, 0.f, 0.f, 0.f);
    if (i < N) deg[i] = 1.0f;
}

// ---------------------------------------------------------------------------
// 2) degree: one thread per edge, atomic +1 at destination node
// ---------------------------------------------------------------------------
__global__ __launch_bounds__(256) void gcn_degree(const int* __restrict__ ei,
                                                  float* __restrict__ deg, int E) {
    int e = blockIdx.x * 256 + threadIdx.x;
    if (e < E) atomAddF(&deg[ei[E + e]], 1.0f);
}

// ---------------------------------------------------------------------------
// 3) GEMM xw = x @ W, fp32 WMMA 16x16x4.
//    One block = one 16-row tile of x (staged in LDS, 8KB, shared by 8 waves).
//    Wave w computes the 16x16 tile at columns [16w, 16w+16).
//    N is a multiple of 16 (100000 = 6250*16).
// ---------------------------------------------------------------------------
__global__ __launch_bounds__(256) void gcn_gemm(const float* __restrict__ x,
                                                const float* __restrict__ W,
                                                float* __restrict__ xw, int N) {
    __shared__ float lds_x[16 * D_FEAT];   // 8 KB A-tile
    const int tid = threadIdx.x;
    const int rowBase = blockIdx.x * 16;

    // cooperative coalesced load: 2048 floats = 256 threads * 2 float4
    {
        const float4* src = (const float4*)(x + (size_t)rowBase * D_FEAT);
        float4* dst = (float4*)lds_x;
        dst[tid]       = src[tid];
        dst[tid + 256] = src[tid + 256];
    }
    __syncthreads();

    const int wave = tid >> 5;          // 0..7 -> output column tile
    const int lane = tid & 31;
    const int m    = lane & 15;         // M index (A rows / D columns-in-lane)
    const int hi   = lane >> 4;         // half-wave selector
    const int n    = (wave << 4) + m;   // global output column

    const float* arow = lds_x + m * D_FEAT;
    v8f acc = {};

    #pragma unroll
    for (int k0 = 0; k0 < D_FEAT; k0 += 4) {
        const int ka = k0 + (hi << 1);  // lanes 0-15: K{k0,k0+1}; 16-31: K{k0+2,k0+3}
        v2f a, b;
        a.x = arow[ka];
        a.y = arow[ka + 1];
        b.x = W[(size_t)ka * D_FEAT + n];
        b.y = W[(size_t)(ka + 1) * D_FEAT + n];
        // D = A(16x4, f32) * B(4x16, f32) + C ; exact fp32 accumulate
        acc = __builtin_amdgcn_wmma_f32_16x16x4_f32(
            /*neg_a=*/false, a, /*neg_b=*/false, b,
            /*c_mod=*/(short)0, acc, /*reuse_a=*/false, /*reuse_b=*/false);
    }

    // C/D layout: VGPR r -> M = r + 8*hi, N = lane&15
    float* orow = xw + (size_t)(rowBase + (hi << 3)) * D_FEAT + (wave << 4) + m;
    #pragma unroll
    for (int r = 0; r < 8; ++r) {
        orow[(size_t)r * D_FEAT] = acc[r];
    }
}

// ---------------------------------------------------------------------------
// 4) dis = rsqrt(deg) in place (deg >= 1 always)
// ---------------------------------------------------------------------------
__global__ __launch_bounds__(256) void gcn_rsqrt(float* __restrict__ deg, int N) {
    int i = blockIdx.x * 256 + threadIdx.x;
    if (i < N) deg[i] = rsqrtf(deg[i]);
}

// ---------------------------------------------------------------------------
// 5) edge gather-scale-scatter: one wave32 per edge, 4 feats/lane.
//    xw (51MB) and agg (51MB) both fit in the 192MB L2 -> atomics L2-resident.
// ---------------------------------------------------------------------------
__global__ __launch_bounds__(256) void gcn_edge_agg(const int* __restrict__ ei,
                                                    const float* __restrict__ xw,
                                                    const float* __restrict__ dis,
                                                    float* __restrict__ agg, int E) {
    const int e = (blockIdx.x << 3) + (threadIdx.x >> 5);
    if (e >= E) return;
    const int lane = threadIdx.x & 31;
    const int r = ei[e];          // source
    const int c = ei[E + e];      // destination
    const float nrm = dis[r] * dis[c];

    const float4 v = *(const float4*)(xw + (size_t)r * D_FEAT + (lane << 2));
    float* dst = agg + (size_t)c * D_FEAT + (lane << 2);
    atomAddF(dst + 0, v.x * nrm);
    atomAddF(dst + 1, v.y * nrm);
    atomAddF(dst + 2, v.z * nrm);
    atomAddF(dst + 3, v.w * nrm);
}

// ---------------------------------------------------------------------------
// 6) finalize: out = PReLU(agg + xw * dis^2 + b)
// ---------------------------------------------------------------------------
__global__ __launch_bounds__(256) void gcn_finalize(const float4* __restrict__ agg4,
                                                    const float4* __restrict__ xw4,
                                                    const float* __restrict__ dis,
                                                    const float* __restrict__ b,
                                                    const float* __restrict__ alpha,
                                                    float4* __restrict__ out4, int N) {
    int idx = blockIdx.x * 256 + threadIdx.x;        // one per (node, 4-feature group)
    int n4 = N * (D_FEAT / 4);
    if (idx >= n4) return;
    const int node = idx >> 5;                       // 32 groups per node
    const int g    = (idx & 31) << 2;                // feature offset 0..124
    const float ds = dis[node];
    const float d2 = ds * ds;

    const float4 a  = agg4[idx];
    const float4 w  = xw4[idx];
    const float4 bb = *(const float4*)(b + g);
    const float4 al = *(const float4*)(alpha + g);

    float o0 = a.x + w.x * d2 + bb.x;
    float o1 = a.y + w.y * d2 + bb.y;
    float o2 = a.z + w.z * d2 + bb.z;
    float o3 = a.w + w.w * d2 + bb.w;
    o0 = (o0 >= 0.f) ? o0 : al.x * o0;
    o1 = (o1 >= 0.f) ? o1 : al.y * o1;
    o2 = (o2 >= 0.f) ? o2 : al.z * o2;
    o3 = (o3 >= 0.f) ? o3 : al.w * o3;
    out4[idx] = make_float4(o0, o1, o2, o3);
}

// ---------------------------------------------------------------------------
// launch
// ---------------------------------------------------------------------------
extern "C" void kernel_launch(void* const* d_in, const int* in_sizes, int n_in,
                              void* d_out, int out_size, void* d_ws, size_t ws_size,
                              hipStream_t stream) {
    const float* x     = (const float*)d_in[0];
    const int*   ei    = (const int*)d_in[1];     // [2, E] flat: row then col
    const float* W     = (const float*)d_in[2];
    const float* b     = (const float*)d_in[3];
    const float* alpha = (const float*)d_in[4];
    float*       out   = (float*)d_out;

    const int N = in_sizes[0] / D_FEAT;
    const int E = in_sizes[1] / 2;

    // workspace: agg [N*128] | xw [N*128] | deg/dis [N]
    float* agg = (float*)d_ws;
    float* xw  = agg + (size_t)N * D_FEAT;
    float* deg = xw + (size_t)N * D_FEAT;

    const int n4 = N * (D_FEAT / 4);

    gcn_init    <<<(n4 + 255) / 256, 256, 0, stream>>>((float4*)agg, deg, N);
    gcn_degree  <<<(E + 255) / 256, 256, 0, stream>>>(ei, deg, E);
    gcn_gemm    <<<N / 16,           256, 0, stream>>>(x, W, xw, N);
    gcn_rsqrt   <<<(N + 255) / 256,  256, 0, stream>>>(deg, N);
    gcn_edge_agg<<<(E + 7) / 8,      256, 0, stream>>>(ei, xw, deg, agg, E);
    gcn_finalize<<<(n4 + 255) / 256, 256, 0, stream>>>((const float4*)agg, (const float4*)xw,
                                                       deg, b, alpha, (float4*)out, N);
}